// EdgeAwareAttention_54786602828359
// MI455X (gfx1250) — compile-verified
//
#include <hip/hip_runtime.h>
#include <hip/hip_bf16.h>

#define DIM 64
#define NUM_HEADS 4
#define HEAD_DIM 16
#define SCALE_F 0.25f            // HEAD_DIM^-0.5
#define NEG_INF_ORD 0x807FFFFF   // f2ord(-inf)

typedef __attribute__((ext_vector_type(2))) float v2f;
typedef __attribute__((ext_vector_type(8))) float v8f;
typedef __attribute__((ext_vector_type(4))) unsigned int v4u;
typedef __attribute__((ext_vector_type(8))) int v8i_t;
typedef __attribute__((ext_vector_type(4))) int v4i_t;

#if __has_builtin(__builtin_amdgcn_tensor_load_to_lds)
#define HAVE_TDM 1
#else
#define HAVE_TDM 0
#endif

// Monotonic float<->ordered-int mapping so we can use integer atomicMax for
// the segment max (scores may be negative).
__device__ __forceinline__ int f2ord(float f) {
  int b = __float_as_int(f);
  return b >= 0 ? b : (b ^ 0x7fffffff);
}
__device__ __forceinline__ float ord2f(int b) {
  return __int_as_float(b >= 0 ? b : (b ^ 0x7fffffff));
}

#if HAVE_TDM
// ---------------------------------------------------------------------------
// TDM: DMA a 64x64 f32 row-major tile from global into LDS (2-D descriptor).
// D# layout per CDNA5 ISA ch.8:
//  group0: [1:0]=count=1, [63:32]=lds_addr, [120:64]=global_addr, [127:126]=2
//  group1: [17:16]=data_size(2 -> 4B), [79:48]=tensor_dim0, [111:80]=tensor_dim1,
//          [127:112]=tile_dim0, [143:128]=tile_dim1, [207:160]=dim0_stride
//  groups 2/3 (+ trailing group): zero (tensor is 2-D)
// This toolchain exposes the 6-arg builtin:
//  (uint32x4 g0, int32x8 g1, int32x4, int32x4, int32x8, i32 cpol)
// Issued by one wave; completion via TENSORcnt, published with a barrier.
// ---------------------------------------------------------------------------
__device__ __forceinline__ void tdm_load_64x64_f32(unsigned lds_byte,
                                                   const float* gsrc) {
  unsigned long long ga = (unsigned long long)(uintptr_t)gsrc;
  v4u g0;
  g0.x = 1u;                                            // count=1, user D#
  g0.y = lds_byte;                                      // lds_addr
  g0.z = (unsigned)ga;                                  // global_addr[31:0]
  g0.w = (unsigned)((ga >> 32) & 0x01FFFFFFull) | 0x80000000u; // [56:32]|type=2
  v8i_t g1;
  g1[0] = 0x00020000;          // wg_mask=0, data_size=2 (4 bytes)
  g1[1] = (int)(64u << 16);    // tensor_dim0 = 64 (bits 79:48)
  g1[2] = (int)(64u << 16);    // tensor_dim0 hi = 0, tensor_dim1 = 64 (111:80)
  g1[3] = (int)(64u << 16);    // tensor_dim1 hi = 0, tile_dim0 = 64 (127:112)
  g1[4] = 64;                  // tile_dim1 = 64, tile_dim2 = 0
  g1[5] = 64;                  // tensor_dim0_stride = 64 (bits 207:160)
  g1[6] = 0;
  g1[7] = 0;
  v4i_t z4 = {0, 0, 0, 0};
  v8i_t z8 = {0, 0, 0, 0, 0, 0, 0, 0};
  __builtin_amdgcn_tensor_load_to_lds(g0, g1, z4, z4, z8, 0);
}
#endif

// ---------------------------------------------------------------------------
// WMMA fp32 16x16x4 tile GEMM:  Y[m0:m0+16, :] = X[m0:m0+16, :] @ W^T + bias
// A-frag (32-bit A 16x4, ISA 7.12.2): lane L holds row M=L&15,
//   VGPR0 = K = 2*(L>>4), VGPR1 = K = 2*(L>>4)+1.
// B-frag (4x16, row striped): lane L holds col N=L&15,
//   b.x = B[K=2*(L>>4)][N], b.y = B[K=2*(L>>4)+1][N], with B = W^T.
// C/D (16x16 f32): VGPR v: lanes 0-15 -> M=v, lanes 16-31 -> M=v+8, N=L&15.
// ---------------------------------------------------------------------------
__device__ __forceinline__ void load_a_frags(v2f a[16], const float* __restrict__ X,
                                             int m0, int lane) {
  const int mr = lane & 15;
  const int kh = lane >> 4;
  const float* xp = X + (size_t)(m0 + mr) * DIM + 2 * kh;
#pragma unroll
  for (int kc = 0; kc < 16; ++kc)
    a[kc] = *(const v2f*)(xp + 4 * kc);
}

// W may point into LDS (addrspace inferred after inlining -> ds_load_b64).
// All 16 B-fragments are preloaded so loads pipeline ahead of the WMMA chain.
__device__ __forceinline__ void wmma_tile(const v2f a[16], const float* W,
                                          const float* __restrict__ bias,
                                          float* __restrict__ Y, int m0, int lane) {
  const int mr = lane & 15;
  const int kh = lane >> 4;
#pragma unroll
  for (int nb = 0; nb < 4; ++nb) {
    // B[k][n] = W^T[k][n] = W[n][k] ; n = nb*16 + mr ; k = 4*kc + 2*kh + {0,1}
    const float* wp = W + (size_t)(nb * 16 + mr) * DIM + 2 * kh;
    v2f b[16];
#pragma unroll
    for (int kc = 0; kc < 16; ++kc)
      b[kc] = *(const v2f*)(wp + 4 * kc);
    v8f acc = {};
#pragma unroll
    for (int kc = 0; kc < 16; ++kc)
      acc = __builtin_amdgcn_wmma_f32_16x16x4_f32(false, a[kc], false, b[kc],
                                                  (short)0, acc, false, false);
    const float bb = bias[nb * 16 + mr];
    float* yp = Y + (size_t)(m0 + 8 * kh) * DIM + nb * 16 + mr;
#pragma unroll
    for (int v = 0; v < 8; ++v)
      yp[(size_t)v * DIM] = acc[v] + bb;   // row m0 + 8*kh + v, col nb*16+mr
  }
}

// ---- fused QKV projection: 8 waves/block, one 16-row tile per wave ---------
__global__ void __launch_bounds__(256)
qkv_proj_kernel(const float* __restrict__ x,
                const float* __restrict__ Wq, const float* __restrict__ bq,
                const float* __restrict__ Wk, const float* __restrict__ bk,
                const float* __restrict__ Wv, const float* __restrict__ bv,
                float* __restrict__ Q, float* __restrict__ K,
                float* __restrict__ V, int ntiles) {
  __shared__ float ldsW[3 * DIM * DIM];   // 48 KB: Wq | Wk | Wv
  const int wid  = (int)(threadIdx.x >> 5);
  const int lane = (int)(threadIdx.x & 31);

#if HAVE_TDM
  if (wid == 0) {                          // one wave drives the TDM
    const unsigned base = (unsigned)(uintptr_t)(void*)ldsW;
    tdm_load_64x64_f32(base, Wq);
    tdm_load_64x64_f32(base + 1u * DIM * DIM * 4u, Wk);
    tdm_load_64x64_f32(base + 2u * DIM * DIM * 4u, Wv);
    __builtin_amdgcn_s_wait_tensorcnt(0);
  }
#else
  for (int i = threadIdx.x; i < DIM * DIM; i += blockDim.x) {
    ldsW[i]               = Wq[i];
    ldsW[DIM * DIM + i]   = Wk[i];
    ldsW[2 * DIM * DIM + i] = Wv[i];
  }
#endif
  __syncthreads();

  const int tile = (int)blockIdx.x * 8 + wid;
  if (tile < ntiles) {                     // wave-uniform: EXEC all-ones
    const int m0 = tile * 16;
    v2f a[16];
    load_a_frags(a, x, m0, lane);
    wmma_tile(a, ldsW,                 bq, Q, m0, lane);
    wmma_tile(a, ldsW + DIM * DIM,     bk, K, m0, lane);
    wmma_tile(a, ldsW + 2 * DIM * DIM, bv, V, m0, lane);
  }
}

// ---- output projection: out = accum @ Wo^T + bo ----------------------------
__global__ void __launch_bounds__(256)
out_proj_kernel(const float* __restrict__ accum,
                const float* __restrict__ Wo, const float* __restrict__ bo,
                float* __restrict__ out, int ntiles) {
  __shared__ float ldsW[DIM * DIM];        // 16 KB
  const int wid  = (int)(threadIdx.x >> 5);
  const int lane = (int)(threadIdx.x & 31);

#if HAVE_TDM
  if (wid == 0) {
    tdm_load_64x64_f32((unsigned)(uintptr_t)(void*)ldsW, Wo);
    __builtin_amdgcn_s_wait_tensorcnt(0);
  }
#else
  for (int i = threadIdx.x; i < DIM * DIM; i += blockDim.x) ldsW[i] = Wo[i];
#endif
  __syncthreads();

  const int tile = (int)blockIdx.x * 8 + wid;
  if (tile < ntiles) {
    const int m0 = tile * 16;
    v2f a[16];
    load_a_frags(a, accum, m0, lane);
    wmma_tile(a, ldsW, bo, out, m0, lane);
  }
}

// ---- scalar tail for N % 16 rows (launches 0 blocks for N = 100000) --------
__global__ void gemm_tail_kernel(const float* __restrict__ X, const float* __restrict__ W,
                                 const float* __restrict__ b, float* __restrict__ Y,
                                 int row0, int nrows) {
  const int idx = blockIdx.x * blockDim.x + threadIdx.x;
  if (idx >= nrows * DIM) return;
  const int r = row0 + idx / DIM;
  const int c = idx % DIM;
  float s = b[c];
#pragma unroll 8
  for (int i = 0; i < DIM; ++i) s += X[(size_t)r * DIM + i] * W[(size_t)c * DIM + i];
  Y[(size_t)r * DIM + c] = s;
}

// ---- init: seg_max = ord(-inf), seg_sum = 0, accum = 0 ---------------------
__global__ void init_kernel(int* __restrict__ segmax, float* __restrict__ segsum,
                            float* __restrict__ accum, int N_) {
  const int i = blockIdx.x * blockDim.x + threadIdx.x;
  if (i < N_ * NUM_HEADS) { segmax[i] = NEG_INF_ORD; segsum[i] = 0.0f; }
  if (i < N_ * DIM) accum[i] = 0.0f;
}

// ---- pass 1: per-edge scores + segment max (ordered-int atomicMax) ---------
// 16 lanes per edge; lane l holds dims 4l..4l+3 (head = l>>2) as one float4.
__global__ void score_kernel(const float* __restrict__ Q, const float* __restrict__ K,
                             const int* __restrict__ ei, const int* __restrict__ etype,
                             const float* __restrict__ ebias,
                             float* __restrict__ scores, int* __restrict__ segmax,
                             int E_) {
  const int gid = blockIdx.x * blockDim.x + threadIdx.x;
  const int edge = gid >> 4;
  const int l = gid & 15;
  if (edge >= E_) return;
  const int src = ei[edge];
  const int tgt = ei[E_ + edge];
  const float4 q = ((const float4*)(Q + (size_t)tgt * DIM))[l];
  const float4 k = ((const float4*)(K + (size_t)src * DIM))[l];
  float part = q.x * k.x + q.y * k.y + q.z * k.z + q.w * k.w;
  part += __shfl_xor(part, 1, 32);      // reduce 4 lanes of one head
  part += __shfl_xor(part, 2, 32);
  if ((l & 3) == 0) {
    const int h = l >> 2;
    const float s = part * SCALE_F + ebias[etype[edge] * NUM_HEADS + h];
    scores[(size_t)edge * NUM_HEADS + h] = s;
    atomicMax(&segmax[tgt * NUM_HEADS + h], f2ord(s));
  }
}

// ---- pass 2: exp(s - max) + segment sum; one thread per (edge, head) -------
__global__ void expsum_kernel(float* __restrict__ scores, const int* __restrict__ segmax,
                              const int* __restrict__ ei, float* __restrict__ segsum,
                              int E_) {
  const int gid = blockIdx.x * blockDim.x + threadIdx.x;
  const int edge = gid >> 2;
  const int h = gid & 3;
  if (edge >= E_) return;
  const int tgt = ei[E_ + edge];
  float m = ord2f(segmax[tgt * NUM_HEADS + h]);
  m = fmaxf(m, -1e9f);                  // matches reference empty-segment clamp
  const float es = __expf(scores[(size_t)edge * NUM_HEADS + h] - m);
  scores[(size_t)edge * NUM_HEADS + h] = es;   // in-place: now exp-scores
  atomicAdd(&segsum[tgt * NUM_HEADS + h], es);
}

// ---- pass 3: weighted scatter-add of V[src] into accum[tgt] ----------------
__global__ void aggregate_kernel(const float* __restrict__ V, const float* __restrict__ scores,
                                 const float* __restrict__ segsum, const int* __restrict__ ei,
                                 float* __restrict__ accum, int E_) {
  const int gid = blockIdx.x * blockDim.x + threadIdx.x;
  const int edge = gid >> 4;
  const int l = gid & 15;
  if (edge >= E_) return;
  const int src = ei[edge];
  const int tgt = ei[E_ + edge];
  const int h = l >> 2;
  const float w = scores[(size_t)edge * NUM_HEADS + h] /
                  (segsum[tgt * NUM_HEADS + h] + 1e-12f);
  const float4 v = ((const float4*)(V + (size_t)src * DIM))[l];
  float* o = accum + (size_t)tgt * DIM + l * 4;
  atomicAdd(o + 0, v.x * w);
  atomicAdd(o + 1, v.y * w);
  atomicAdd(o + 2, v.z * w);
  atomicAdd(o + 3, v.w * w);
}

extern "C" void kernel_launch(void* const* d_in, const int* in_sizes, int n_in,
                              void* d_out, int out_size, void* d_ws, size_t ws_size,
                              hipStream_t stream) {
  const float* x     = (const float*)d_in[0];
  const int*   ei    = (const int*)d_in[1];    // [2, E] row-major
  const int*   etype = (const int*)d_in[2];    // [E]
  const float* Wq = (const float*)d_in[4];  const float* bq = (const float*)d_in[5];
  const float* Wk = (const float*)d_in[6];  const float* bk = (const float*)d_in[7];
  const float* Wv = (const float*)d_in[8];  const float* bv = (const float*)d_in[9];
  const float* Wo = (const float*)d_in[10]; const float* bo = (const float*)d_in[11];
  const float* ebias = (const float*)d_in[12];
  float* out = (float*)d_out;

  const int N_ = in_sizes[0] / DIM;
  const int E_ = in_sizes[2];

  // Workspace carve-up (~126 MB for N=100k, E=1.25M)
  char* ws = (char*)d_ws;
  size_t off = 0;
  float* Q      = (float*)(ws + off); off += (size_t)N_ * DIM * 4;
  float* K      = (float*)(ws + off); off += (size_t)N_ * DIM * 4;
  float* V      = (float*)(ws + off); off += (size_t)N_ * DIM * 4;
  float* accum  = (float*)(ws + off); off += (size_t)N_ * DIM * 4;
  float* scores = (float*)(ws + off); off += (size_t)E_ * NUM_HEADS * 4;
  float* segsum = (float*)(ws + off); off += (size_t)N_ * NUM_HEADS * 4;
  int*   segmax = (int*)  (ws + off); off += (size_t)N_ * NUM_HEADS * 4;
  (void)ws_size;

  const int ntiles = N_ / 16;
  const int tail0  = ntiles * 16;
  const int tail_n = N_ - tail0;

  // init segmax/segsum/accum
  {
    const size_t tot = (size_t)N_ * DIM;
    init_kernel<<<(unsigned)((tot + 255) / 256), 256, 0, stream>>>(segmax, segsum, accum, N_);
  }
  // fused QKV projection (TDM W-staging + WMMA fp32 16x16x4), 8 waves/block
  if (ntiles > 0)
    qkv_proj_kernel<<<(unsigned)((ntiles + 7) / 8), 256, 0, stream>>>(
        x, Wq, bq, Wk, bk, Wv, bv, Q, K, V, ntiles);
  if (tail_n > 0) {
    const unsigned g = (unsigned)((tail_n * DIM + 255) / 256);
    gemm_tail_kernel<<<g, 256, 0, stream>>>(x, Wq, bq, Q, tail0, tail_n);
    gemm_tail_kernel<<<g, 256, 0, stream>>>(x, Wk, bk, K, tail0, tail_n);
    gemm_tail_kernel<<<g, 256, 0, stream>>>(x, Wv, bv, V, tail0, tail_n);
  }
  // edge pass 1: scores + segment max
  {
    const size_t tot = (size_t)E_ * 16;
    score_kernel<<<(unsigned)((tot + 255) / 256), 256, 0, stream>>>(
        Q, K, ei, etype, ebias, scores, segmax, E_);
  }
  // edge pass 2: exp + segment sum
  {
    const size_t tot = (size_t)E_ * NUM_HEADS;
    expsum_kernel<<<(unsigned)((tot + 255) / 256), 256, 0, stream>>>(
        scores, segmax, ei, segsum, E_);
  }
  // edge pass 3: weighted scatter-add
  {
    const size_t tot = (size_t)E_ * 16;
    aggregate_kernel<<<(unsigned)((tot + 255) / 256), 256, 0, stream>>>(
        V, scores, segsum, ei, accum, E_);
  }
  // output projection (TDM W-staging + WMMA fp32 16x16x4)
  if (ntiles > 0)
    out_proj_kernel<<<(unsigned)((ntiles + 7) / 8), 256, 0, stream>>>(
        accum, Wo, bo, out, ntiles);
  if (tail_n > 0) {
    const unsigned g = (unsigned)((tail_n * DIM + 255) / 256);
    gemm_tail_kernel<<<g, 256, 0, stream>>>(accum, Wo, bo, out, tail0, tail_n);
  }
}